// GraphSAGE_13615046328735
// MI455X (gfx1250) — compile-verified
//
#include <hip/hip_runtime.h>

typedef __attribute__((ext_vector_type(2))) float v2f;
typedef __attribute__((ext_vector_type(8))) float v8f;

#define NND 100000
#define NED 1600000
#define INDIM 48
#define HIDD 16
#define NCLS 8

__device__ __forceinline__ float eluf(float v) {
    return v > 0.0f ? v : (__expf(v) - 1.0f);
}

// ---------------- degree ----------------
__global__ void k_degree(const int* __restrict__ dst, float* __restrict__ deg, int E) {
    int i = blockIdx.x * blockDim.x + threadIdx.x;
    if (i < E) atomicAdd(&deg[dst[i]], 1.0f);
}

__global__ void k_invdeg(float* __restrict__ deg, int n) {
    int i = blockIdx.x * blockDim.x + threadIdx.x;
    if (i < n) deg[i] = 1.0f / fmaxf(deg[i], 1.0f);
}

// ---------------- edge scatter-add (mean numerator) ----------------
template <int F>
__global__ void k_scatter(const float* __restrict__ h, const int* __restrict__ src,
                          const int* __restrict__ dst, float* __restrict__ agg,
                          long long total) {
    constexpr int C4 = F / 4;
    long long i = (long long)blockIdx.x * blockDim.x + threadIdx.x;
    if (i >= total) return;
    int e = (int)(i / C4);
    int c = (int)(i % C4) * 4;
    int s = src[e], d = dst[e];
    const float4 v = *(const float4*)(h + (long long)s * F + c);
    float* ap = agg + (long long)d * F + c;
    atomicAdd(ap + 0, v.x);
    atomicAdd(ap + 1, v.y);
    atomicAdd(ap + 2, v.z);
    atomicAdd(ap + 3, v.w);
}

// ---------------- WMMA combine: out = ELU(agg*invdeg @ Wl^T + x @ Wr^T + b) ----------------
// One wave32 computes a 16-node x 16-channel tile via V_WMMA_F32_16X16X4_F32.
template <int F, int H>
__global__ void k_combine(const float* __restrict__ x, const float* __restrict__ agg,
                          const float* __restrict__ invdeg,
                          const float* __restrict__ Wl, const float* __restrict__ Wr,
                          const float* __restrict__ b, float* __restrict__ out,
                          int n_nodes) {
    const int lane = threadIdx.x & 31;
    const int wave = threadIdx.x >> 5;
    const int node0 = (blockIdx.x * (blockDim.x >> 5) + wave) * 16;
    if (node0 >= n_nodes) return;  // wave-uniform: EXEC stays all-1s for WMMA

    const int m = lane & 15;              // A-matrix row (node in tile); also C/D column N
    const int koff = (lane >> 4) * 2;     // K sub-offset for A and B fragments
    const int node = node0 + m;
    const float idg = invdeg[node];
    const bool wv = (m < H);              // valid output channel (pad B rows beyond H with 0)

    v8f acc;
    const float bn = wv ? b[m] : 0.0f;    // bias folded into the accumulator init
#pragma unroll
    for (int r = 0; r < 8; ++r) acc[r] = bn;

    const float* xr = x + (long long)node * F;
    const float* ar = agg + (long long)node * F;
    const float* wlr = wv ? (Wl + m * F) : Wl;  // row m of [H,F]; dummy row if padded
    const float* wrr = wv ? (Wr + m * F) : Wr;

#pragma unroll
    for (int k0 = 0; k0 < F; k0 += 4) {
        v2f a, w;
        // neighbor-mean path: A = (agg * invdeg) tile, B = Wl^T fragment
        a.x = ar[k0 + koff] * idg;
        a.y = ar[k0 + koff + 1] * idg;
        w.x = wv ? wlr[k0 + koff] : 0.0f;
        w.y = wv ? wlr[k0 + koff + 1] : 0.0f;
        acc = __builtin_amdgcn_wmma_f32_16x16x4_f32(false, a, false, w,
                                                    (short)0, acc, false, false);
        // root path: A = x tile, B = Wr^T fragment
        a.x = xr[k0 + koff];
        a.y = xr[k0 + koff + 1];
        w.x = wv ? wrr[k0 + koff] : 0.0f;
        w.y = wv ? wrr[k0 + koff + 1] : 0.0f;
        acc = __builtin_amdgcn_wmma_f32_16x16x4_f32(false, a, false, w,
                                                    (short)0, acc, false, false);
    }

    if (m < H) {
#pragma unroll
        for (int r = 0; r < 8; ++r) {
            int mr = r + (lane >> 4) * 8;  // C/D: VGPR r holds row r (+8 for upper half-wave)
            out[(long long)(node0 + mr) * H + m] = eluf(acc[r]);
        }
    }
}

// ---------------- log_softmax over 8 classes, in place ----------------
__global__ void k_logsoftmax(float* __restrict__ out, int n_nodes) {
    int i = blockIdx.x * blockDim.x + threadIdx.x;
    if (i >= n_nodes) return;
    float v[NCLS];
    float mx = -3.4e38f;
#pragma unroll
    for (int c = 0; c < NCLS; ++c) {
        v[c] = out[i * NCLS + c];
        mx = fmaxf(mx, v[c]);
    }
    float s = 0.0f;
#pragma unroll
    for (int c = 0; c < NCLS; ++c) s += __expf(v[c] - mx);
    float ls = __logf(s);
#pragma unroll
    for (int c = 0; c < NCLS; ++c) out[i * NCLS + c] = v[c] - mx - ls;
}

extern "C" void kernel_launch(void* const* d_in, const int* in_sizes, int n_in,
                              void* d_out, int out_size, void* d_ws, size_t ws_size,
                              hipStream_t stream) {
    const float* x  = (const float*)d_in[0];
    const int* ei   = (const int*)d_in[1];
    const int* src  = ei;
    const int* dst  = ei + NED;
    const float* W1l = (const float*)d_in[2];
    const float* W1r = (const float*)d_in[3];
    const float* b1  = (const float*)d_in[4];
    const float* W2l = (const float*)d_in[5];
    const float* W2r = (const float*)d_in[6];
    const float* b2  = (const float*)d_in[7];
    const float* W3l = (const float*)d_in[8];
    const float* W3r = (const float*)d_in[9];
    const float* b3  = (const float*)d_in[10];
    float* out = (float*)d_out;

    float* ws  = (float*)d_ws;
    float* deg = ws;                                  // N floats (becomes invdeg)
    float* agg = deg + NND;                           // N*48 floats (reused per layer)
    float* h1  = agg + (size_t)NND * INDIM;           // N*16
    float* h2  = h1 + (size_t)NND * HIDD;             // N*16

    const int TB = 256;
    const int combBlocks = (NND + 127) / 128;         // 8 waves/block * 16 nodes/wave

    // degree (identical for all three layers: compute once)
    hipMemsetAsync(deg, 0, sizeof(float) * NND, stream);
    k_degree<<<(NED + TB - 1) / TB, TB, 0, stream>>>(dst, deg, NED);
    k_invdeg<<<(NND + TB - 1) / TB, TB, 0, stream>>>(deg, NND);

    // ---- layer 1: F=48 -> H=16 ----
    {
        long long tot = (long long)NED * (INDIM / 4);
        hipMemsetAsync(agg, 0, sizeof(float) * (size_t)NND * INDIM, stream);
        k_scatter<INDIM><<<(int)((tot + TB - 1) / TB), TB, 0, stream>>>(x, src, dst, agg, tot);
        k_combine<INDIM, HIDD><<<combBlocks, TB, 0, stream>>>(x, agg, deg, W1l, W1r, b1, h1, NND);
    }
    // ---- layer 2: F=16 -> H=16 ----
    {
        long long tot = (long long)NED * (HIDD / 4);
        hipMemsetAsync(agg, 0, sizeof(float) * (size_t)NND * HIDD, stream);
        k_scatter<HIDD><<<(int)((tot + TB - 1) / TB), TB, 0, stream>>>(h1, src, dst, agg, tot);
        k_combine<HIDD, HIDD><<<combBlocks, TB, 0, stream>>>(h1, agg, deg, W2l, W2r, b2, h2, NND);
    }
    // ---- layer 3: F=16 -> H=8, then log_softmax in place ----
    {
        long long tot = (long long)NED * (HIDD / 4);
        hipMemsetAsync(agg, 0, sizeof(float) * (size_t)NND * HIDD, stream);
        k_scatter<HIDD><<<(int)((tot + TB - 1) / TB), TB, 0, stream>>>(h2, src, dst, agg, tot);
        k_combine<HIDD, NCLS><<<combBlocks, TB, 0, stream>>>(h2, agg, deg, W3l, W3r, b3, out, NND);
        k_logsoftmax<<<(NND + TB - 1) / TB, TB, 0, stream>>>(out, NND);
    }
}